// GATTopLayer_81286551044791
// MI455X (gfx1250) — compile-verified
//
#include <hip/hip_runtime.h>
#include <hip/hip_bf16.h>

// Problem constants (from the reference)
#define NN   50000
#define EE   600000
#define INC  128
#define HH   8
#define DD   16
#define HD   128
#define MTILES 3125          // NN / 16 exactly

typedef __attribute__((ext_vector_type(2))) float v2f;
typedef __attribute__((ext_vector_type(8))) float v8f;

// ---------------------------------------------------------------------------
// order-preserving float <-> uint mapping for atomicMax on floats
// ---------------------------------------------------------------------------
__device__ __forceinline__ unsigned fkey(float f) {
  unsigned b = __float_as_uint(f);
  return (b & 0x80000000u) ? ~b : (b | 0x80000000u);
}
__device__ __forceinline__ float funkey(unsigned k) {
  unsigned b = (k & 0x80000000u) ? (k & 0x7FFFFFFFu) : ~k;
  return __uint_as_float(b);
}

// ---------------------------------------------------------------------------
// generic fill
// ---------------------------------------------------------------------------
__global__ void k_fill_u32(unsigned* __restrict__ p, unsigned v, int n) {
  int t = blockIdx.x * blockDim.x + threadIdx.x;
  if (t < n) p[t] = v;
}

// ---------------------------------------------------------------------------
// feat = h @ fc_w.T  via V_WMMA_F32_16X16X4_F32
//
//   A tile: 16x4 of h rows.   Lane<16: row=lane,    K = k0,k0+1
//                             Lane>=16: row=lane-16, K = k0+2,k0+3
//   B tile: 4x16 of fc_w^T (B[k][c] = fc_w[c][k]), mirrored lane layout.
//   D v8f:  VGPR i -> row i (lanes 0-15) / row 8+i (lanes 16-31), col = lane&15
//
// LDS layout groups the 4 K-values of one WMMA step contiguously per column:
//     Bs[(kq*128 + c)*4 + (k&3)] = fc_w[c][k],  kq = k>>2
// so each B fragment is ONE ds_load_b64 at base (lrow*4+koff)*4 bytes with a
// compile-time immediate offset kq*2048 + t*256 (max 65280, fits the 16-bit
// DS offset).  Low lanes hit banks {4c,4c+1}, high lanes {4c+2,4c+3}: the 32
// lanes cover all 64 banks conflict-free.  Staging is one global_load_b128 +
// one ds_store_b128 per step (contiguous across lanes, conflict-free).
// ---------------------------------------------------------------------------
__global__ void k_gemm_feat(const float* __restrict__ h,
                            const float* __restrict__ w,
                            float* __restrict__ feat) {
  __shared__ float Bs[128 * 128];   // 64 KB (WGP allows up to 320 KB)
  const int tid = threadIdx.x;      // 128 threads = 4 waves

  {
    const float4* wr = (const float4*)(w + tid * INC);   // column c = tid
#pragma unroll 4
    for (int kq = 0; kq < 32; ++kq)
      *(float4*)(Bs + kq * 512 + tid * 4) = wr[kq];
  }
  __syncthreads();

  const int wave = tid >> 5;
  const int lane = tid & 31;
  const int tile = blockIdx.x * 4 + wave;     // wave-uniform guard keeps EXEC full
  if (tile < MTILES) {
    const int lrow = lane & 15;
    const int koff = (lane >> 4) << 1;        // 0 for lanes 0-15, 2 for 16-31
    const float* hrow  = h + (size_t)(tile * 16 + lrow) * INC;
    const float* bbase = Bs + lrow * 4 + koff;

    __builtin_prefetch(hrow, 0, 0);           // global_prefetch_b8
    __builtin_prefetch(hrow + 64, 0, 0);

    v8f acc[8] = {};
#pragma unroll
    for (int kq = 0; kq < 32; ++kq) {
      const v2f a = *(const v2f*)(hrow + kq * 4 + koff);     // global_load_b64
#pragma unroll
      for (int t = 0; t < 8; ++t) {
        const v2f b = *(const v2f*)(bbase + kq * 512 + t * 64);  // ds_load_b64, imm offset
        acc[t] = __builtin_amdgcn_wmma_f32_16x16x4_f32(
            false, a, false, b, (short)0, acc[t], false, false);
      }
    }

    const int mbase = tile * 16 + ((lane >> 4) << 3);
#pragma unroll
    for (int t = 0; t < 8; ++t)
#pragma unroll
      for (int i = 0; i < 8; ++i)
        feat[(size_t)(mbase + i) * HD + t * 16 + lrow] = acc[t][i];
  }
}

// ---------------------------------------------------------------------------
// el/er per (node, head)
// ---------------------------------------------------------------------------
__global__ void k_elr(const float* __restrict__ feat,
                      const float* __restrict__ al,
                      const float* __restrict__ ar,
                      float* __restrict__ el, float* __restrict__ er) {
  int t = blockIdx.x * blockDim.x + threadIdx.x;
  if (t >= NN * HH) return;
  const int n = t >> 3, hh = t & 7;
  const float* f = feat + n * HD + hh * DD;
  const float* L = al + hh * DD;
  const float* R = ar + hh * DD;
  float sl = 0.f, sr = 0.f;
#pragma unroll
  for (int d = 0; d < DD; ++d) { sl += f[d] * L[d]; sr += f[d] * R[d]; }
  el[t] = sl; er[t] = sr;
}

// ---------------------------------------------------------------------------
// edge pass 1: e = leaky_relu(el[src]+er[dst]); segment max into mkey[dst]
// ---------------------------------------------------------------------------
__global__ void k_edge1(const int* __restrict__ src, const int* __restrict__ dst,
                        const float* __restrict__ el, const float* __restrict__ er,
                        float* __restrict__ e_ws, unsigned* __restrict__ mkey) {
  int t = blockIdx.x * blockDim.x + threadIdx.x;
  if (t >= EE * HH) return;
  const int e = t >> 3, hh = t & 7;
  const int s = src[e], d = dst[e];
  float x = el[s * HH + hh] + er[d * HH + hh];
  x = x > 0.f ? x : 0.2f * x;
  e_ws[t] = x;
  atomicMax(&mkey[d * HH + hh], fkey(x));
}

// ---------------------------------------------------------------------------
// edge pass 2: ex = exp(e - m[dst]); segment sum into z[dst]
// ---------------------------------------------------------------------------
__global__ void k_edge2(const int* __restrict__ dst,
                        const unsigned* __restrict__ mkey,
                        float* __restrict__ e_ws, float* __restrict__ z) {
  int t = blockIdx.x * blockDim.x + threadIdx.x;
  if (t >= EE * HH) return;
  const int e = t >> 3, hh = t & 7;
  const int d = dst[e];
  const float ex = __expf(e_ws[t] - funkey(mkey[d * HH + hh]));
  e_ws[t] = ex;
  atomicAdd(&z[d * HH + hh], ex);
}

// ---------------------------------------------------------------------------
// edge pass 3: a = ex/z[dst]; attn out; scatter feat[src]*a into h_out[dst];
// 8x8 edge MLP -> val = 1-w; scatter val into sum_w[src] and sum_w[dst]
// ---------------------------------------------------------------------------
__global__ void k_edge3(const int* __restrict__ src, const int* __restrict__ dst,
                        const float* __restrict__ feat,
                        const float* __restrict__ e_ws, const float* __restrict__ z,
                        const float* __restrict__ aw1, const float* __restrict__ ab1,
                        const float* __restrict__ aw2, const float* __restrict__ ab2,
                        float* __restrict__ attn_out, float* __restrict__ out_h,
                        float* __restrict__ sum_w) {
  int t = blockIdx.x * blockDim.x + threadIdx.x;
  if (t >= EE * HH) return;
  const int e = t >> 3, hh = t & 7;
  const int s = src[e], d = dst[e];

  const float a = e_ws[t] / z[d * HH + hh];
  attn_out[t] = a;

  const float* fs = feat + s * HD + hh * DD;
  float* od = out_h + d * HD + hh * DD;
#pragma unroll
  for (int dd = 0; dd < DD; ++dd) atomicAdd(&od[dd], fs[dd] * a);

  float av[HH];
#pragma unroll
  for (int k = 0; k < HH; ++k) av[k] = e_ws[e * HH + k] / z[d * HH + k];

  float acc2 = ab2[hh];
#pragma unroll
  for (int j = 0; j < HH; ++j) {
    float hid = ab1[j];
#pragma unroll
    for (int k = 0; k < HH; ++k) hid += av[k] * aw1[j * HH + k];
    hid = hid > 0.f ? hid : 0.f;
    acc2 += hid * aw2[hh * HH + j];
  }
  const float val = 1.f - acc2;
  atomicAdd(&sum_w[s * HH + hh], val);
  atomicAdd(&sum_w[d * HH + hh], val);
}

// ---------------------------------------------------------------------------
// node pass: h_out += gat_bias + MLP(sum_w)
// ---------------------------------------------------------------------------
__global__ void k_node(const float* __restrict__ sum_w,
                       const float* __restrict__ tw1, const float* __restrict__ tb1,
                       const float* __restrict__ tw2, const float* __restrict__ tb2,
                       const float* __restrict__ gbias,
                       float* __restrict__ out_h) {
  int t = blockIdx.x * blockDim.x + threadIdx.x;
  if (t >= NN * HD) return;
  const int n = t >> 7, c = t & 127;
  const float* sw = sum_w + n * HH;
  float emb = tb2[c];
#pragma unroll
  for (int j = 0; j < HH; ++j) {
    float hid = tb1[j];
#pragma unroll
    for (int k = 0; k < HH; ++k) hid += sw[k] * tw1[j * HH + k];
    hid = hid > 0.f ? hid : 0.f;
    emb += hid * tw2[c * HH + j];
  }
  out_h[t] += gbias[c] + emb;
}

// ---------------------------------------------------------------------------
// per-column mean/var (128 blocks, one column each)
// ---------------------------------------------------------------------------
__global__ void k_stats(const float* __restrict__ out_h,
                        float* __restrict__ mu, float* __restrict__ var) {
  __shared__ float ss[256], sq[256];
  const int c = blockIdx.x;
  float s = 0.f, q = 0.f;
  for (int n = threadIdx.x; n < NN; n += 256) {
    const float x = out_h[n * HD + c];
    s += x; q += x * x;
  }
  ss[threadIdx.x] = s; sq[threadIdx.x] = q;
  __syncthreads();
  for (int o = 128; o > 0; o >>= 1) {
    if (threadIdx.x < o) {
      ss[threadIdx.x] += ss[threadIdx.x + o];
      sq[threadIdx.x] += sq[threadIdx.x + o];
    }
    __syncthreads();
  }
  if (threadIdx.x == 0) {
    const float m = ss[0] / (float)NN;
    mu[c] = m;
    var[c] = sq[0] / (float)NN - m * m;
  }
}

// ---------------------------------------------------------------------------
// batch-norm + ELU + residual
// ---------------------------------------------------------------------------
__global__ void k_final(const float* __restrict__ h_in,
                        const float* __restrict__ mu, const float* __restrict__ var,
                        const float* __restrict__ gamma, const float* __restrict__ beta,
                        float* __restrict__ out_h) {
  int t = blockIdx.x * blockDim.x + threadIdx.x;
  if (t >= NN * HD) return;
  const int c = t & 127;
  float y = (out_h[t] - mu[c]) * rsqrtf(var[c] + 1e-5f) * gamma[c] + beta[c];
  y = y > 0.f ? y : (__expf(y) - 1.f);
  out_h[t] = h_in[t] + y;
}

// ---------------------------------------------------------------------------
// launcher
// ---------------------------------------------------------------------------
extern "C" void kernel_launch(void* const* d_in, const int* in_sizes, int n_in,
                              void* d_out, int out_size, void* d_ws, size_t ws_size,
                              hipStream_t stream) {
  const float* h       = (const float*)d_in[0];
  const int*   esrc    = (const int*)  d_in[1];
  const int*   edst    = (const int*)  d_in[2];
  const float* fc_w    = (const float*)d_in[4];
  const float* attn_l  = (const float*)d_in[5];
  const float* attn_r  = (const float*)d_in[6];
  const float* gbias   = (const float*)d_in[7];
  const float* aw1     = (const float*)d_in[8];
  const float* ab1     = (const float*)d_in[9];
  const float* aw2     = (const float*)d_in[10];
  const float* ab2     = (const float*)d_in[11];
  const float* tw1     = (const float*)d_in[12];
  const float* tb1     = (const float*)d_in[13];
  const float* tw2     = (const float*)d_in[14];
  const float* tb2     = (const float*)d_in[15];
  const float* bn_g    = (const float*)d_in[16];
  const float* bn_b    = (const float*)d_in[17];

  float* out = (float*)d_out;
  float* out_h    = out;                        // N*HD
  float* out_attn = out + (size_t)NN * HD;      // E*HH
  float* out_tail = out + (size_t)NN * HD + (size_t)EE * HH;  // 512 zeros

  // workspace layout (4-byte units); ~53 MB
  float*    ws    = (float*)d_ws;
  float*    feat  = ws;                                    // 6,400,000
  float*    el    = ws + 6400000;                          //   400,000
  float*    er    = ws + 6800000;                          //   400,000
  float*    e_ws  = ws + 7200000;                          // 4,800,000
  unsigned* mkey  = (unsigned*)(ws + 12000000);            //   400,000
  float*    z     = ws + 12400000;                         //   400,000
  float*    sum_w = ws + 12800000;                         //   400,000
  float*    mu    = ws + 13200000;                         //       128
  float*    var   = ws + 13200128;                         //       128

  const int B = 256;
  // init accumulators / zero outputs (harness poisons with 0xAA)
  k_fill_u32<<<(NN * HD + B - 1) / B, B, 0, stream>>>((unsigned*)out_h, 0u, NN * HD);
  k_fill_u32<<<(NN * HH + B - 1) / B, B, 0, stream>>>(mkey, 0x007FFFFFu /*fkey(-inf)*/, NN * HH);
  k_fill_u32<<<(NN * HH + B - 1) / B, B, 0, stream>>>((unsigned*)z, 0u, NN * HH);
  k_fill_u32<<<(NN * HH + B - 1) / B, B, 0, stream>>>((unsigned*)sum_w, 0u, NN * HH);
  k_fill_u32<<<(512 + B - 1) / B, B, 0, stream>>>((unsigned*)out_tail, 0u, 512);

  // WMMA GEMM: 4 waves/block, one 16-row tile per wave
  k_gemm_feat<<<(MTILES + 3) / 4, 128, 0, stream>>>(h, fc_w, feat);

  k_elr<<<(NN * HH + B - 1) / B, B, 0, stream>>>(feat, attn_l, attn_r, el, er);

  const int EG = (EE * HH + B - 1) / B;
  k_edge1<<<EG, B, 0, stream>>>(esrc, edst, el, er, e_ws, mkey);
  k_edge2<<<EG, B, 0, stream>>>(edst, mkey, e_ws, z);
  k_edge3<<<EG, B, 0, stream>>>(esrc, edst, feat, e_ws, z,
                                aw1, ab1, aw2, ab2, out_attn, out_h, sum_w);

  k_node<<<(NN * HD + B - 1) / B, B, 0, stream>>>(sum_w, tw1, tb1, tw2, tb2, gbias, out_h);
  k_stats<<<HD, 256, 0, stream>>>(out_h, mu, var);
  k_final<<<(NN * HD + B - 1) / B, B, 0, stream>>>(h, mu, var, bn_g, bn_b, out_h);
}